// MultiLayerTetra_69664369541554
// MI455X (gfx1250) — compile-verified
//
#include <hip/hip_runtime.h>
#include <hip/hip_bf16.h>

#define FEAT 32

#ifndef __has_builtin
#define __has_builtin(x) 0
#endif

#if __has_builtin(__builtin_amdgcn_global_load_async_to_lds_b128)
#define ASYNC_LDS 1
#else
#define ASYNC_LDS 0
#endif

typedef int v4i __attribute__((ext_vector_type(4)));
typedef __attribute__((address_space(1))) v4i* gv4i_p;  // global
typedef __attribute__((address_space(3))) v4i* lv4i_p;  // LDS

__device__ __forceinline__ void wait_async0() {
#if __has_builtin(__builtin_amdgcn_s_wait_asynccnt)
  __builtin_amdgcn_s_wait_asynccnt(0);
#elif ASYNC_LDS
  asm volatile("s_wait_asynccnt 0" ::: "memory");
#endif
}

__device__ __forceinline__ float rl_f(float v, int s) {
  return __int_as_float(__builtin_amdgcn_readlane(__float_as_int(v), s));
}

// Barycentric coords of p wrt tetra (v0..v3): solve [e1 e2 e3] l = p - v0 by Cramer,
// w = (1-l1-l2-l3, l1, l2, l3). Matches solve(A,b) with A = [[x],[y],[z],[1,1,1,1]].
__device__ __forceinline__ void bary_from_lds(const float* __restrict__ s_pxyz,
                                              const int pid[4],
                                              float px, float py, float pz,
                                              float w[4]) {
  float vx[4], vy[4], vz[4];
#pragma unroll
  for (int k = 0; k < 4; ++k) {
    const int b = 3 * pid[k];
    vx[k] = s_pxyz[b + 0];
    vy[k] = s_pxyz[b + 1];
    vz[k] = s_pxyz[b + 2];
  }
  const float e1x = vx[1] - vx[0], e1y = vy[1] - vy[0], e1z = vz[1] - vz[0];
  const float e2x = vx[2] - vx[0], e2y = vy[2] - vy[0], e2z = vz[2] - vz[0];
  const float e3x = vx[3] - vx[0], e3y = vy[3] - vy[0], e3z = vz[3] - vz[0];
  const float rx = px - vx[0], ry = py - vy[0], rz = pz - vz[0];

  const float c23x = e2y * e3z - e2z * e3y;
  const float c23y = e2z * e3x - e2x * e3z;
  const float c23z = e2x * e3y - e2y * e3x;
  const float det = e1x * c23x + e1y * c23y + e1z * c23z;
  const float inv = 1.0f / det;

  const float l1 = (rx * c23x + ry * c23y + rz * c23z) * inv;

  const float cr3x = ry * e3z - rz * e3y;
  const float cr3y = rz * e3x - rx * e3z;
  const float cr3z = rx * e3y - ry * e3x;
  const float l2 = (e1x * cr3x + e1y * cr3y + e1z * cr3z) * inv;

  const float c2rx = e2y * rz - e2z * ry;
  const float c2ry = e2z * rx - e2x * rz;
  const float c2rz = e2x * ry - e2y * rx;
  const float l3 = (e1x * c2rx + e1y * c2ry + e1z * c2rz) * inv;

  w[0] = 1.0f - l1 - l2 - l3;
  w[1] = l1;
  w[2] = l2;
  w[3] = l3;
}

__global__ __launch_bounds__(256)
void MultiLayerTetra_69664369541554_kernel(
    const float* __restrict__ xyz,
    const int* __restrict__ cell_id,
    const int* __restrict__ old_cell_id,
    const int* __restrict__ point_index,
    const float* __restrict__ point_xyz,
    const float* __restrict__ field,
    const float* __restrict__ ili,
    const int* __restrict__ thresh_p,
    float* __restrict__ out_feat,
    float* __restrict__ out_imp,
    int Ntot, int P3) {
  extern __shared__ float s_pxyz[];
  const int tid = threadIdx.x;

  // ---- Stage point_xyz (P*3 floats, ~117KB) into LDS via CDNA5 async DMA ----
#if ASYNC_LDS
  {
    const int nvec = P3 >> 2;  // 16B chunks
    for (int i = tid; i < nvec; i += blockDim.x) {
      __builtin_amdgcn_global_load_async_to_lds_b128(
          (gv4i_p)(point_xyz + 4 * i),
          (lv4i_p)(s_pxyz + 4 * i),
          0, 0);
    }
    for (int i = (nvec << 2) + tid; i < P3; i += blockDim.x)
      s_pxyz[i] = point_xyz[i];
    wait_async0();
  }
#else
  for (int i = tid; i < P3; i += blockDim.x) s_pxyz[i] = point_xyz[i];
#endif
  __syncthreads();

  const int thresh = thresh_p[0];
  const int lane = tid & 31;
  const int gw = blockIdx.x * (blockDim.x >> 5) + (tid >> 5);
  const int nw = gridDim.x * (blockDim.x >> 5);
  const int4* __restrict__ pi4 = (const int4*)point_index;

  for (int nbase = gw * 32; nbase < Ntot; nbase += nw * 32) {
    // ---------- Phase 1: lane = sample ----------
    const int n = min(nbase + lane, Ntot - 1);

    // Prefetch next iteration's xyz stream chunk.
    const int nnext = nbase + nw * 32;
    if (nnext < Ntot)
      __builtin_prefetch(xyz + (size_t)3 * (nnext + lane), 0, 1);

    const int cid = cell_id[n];
    const int ocid = old_cell_id[n];
    const float px = xyz[3 * n + 0];
    const float py = xyz[3 * n + 1];
    const float pz = xyz[3 * n + 2];

    const int4 np4 = pi4[cid];
    const int4 op4 = pi4[ocid];
    int npid[4] = {np4.x, np4.y, np4.z, np4.w};
    int opid[4] = {op4.x, op4.y, op4.z, op4.w};

    float wn[4], wo[4];
    bary_from_lds(s_pxyz, npid, px, py, pz, wn);
    bary_from_lds(s_pxyz, opid, px, py, pz, wo);

    // importance = max over 6 of inter_level_importance[cell_id]
    const float* ip = ili + (size_t)cid * 6;
    const float m = fmaxf(fmaxf(fmaxf(ip[0], ip[1]), fmaxf(ip[2], ip[3])),
                          fmaxf(ip[4], ip[5]));
    out_imp[n] = m;

    // ---------- Phase 2: lane = feature (F == 32 == wave32) ----------
    for (int s = 0; s < 32; ++s) {
      const int ns = nbase + s;
      if (ns >= Ntot) break;
      float acc = 0.0f;
#pragma unroll
      for (int k = 0; k < 4; ++k) {  // old cell: no threshold
        const int p = __builtin_amdgcn_readlane(opid[k], s);
        acc = fmaf(rl_f(wo[k], s), field[(size_t)p * FEAT + lane], acc);
      }
#pragma unroll
      for (int k = 0; k < 4; ++k) {  // new cell: rows with pid < thresh are zeroed
        const int p = __builtin_amdgcn_readlane(npid[k], s);
        if (p >= thresh)
          acc = fmaf(rl_f(wn[k], s), field[(size_t)p * FEAT + lane], acc);
      }
      out_feat[(size_t)ns * FEAT + lane] = acc;  // coalesced 128B wave store
    }
  }
}

extern "C" void kernel_launch(void* const* d_in, const int* in_sizes, int n_in,
                              void* d_out, int out_size, void* d_ws, size_t ws_size,
                              hipStream_t stream) {
  const float* xyz = (const float*)d_in[0];
  const int* cell_id = (const int*)d_in[1];
  const int* old_cell_id = (const int*)d_in[2];
  const int* point_index = (const int*)d_in[3];
  const float* point_xyz = (const float*)d_in[4];
  const float* field = (const float*)d_in[5];
  const float* ili = (const float*)d_in[6];
  const int* thresh = (const int*)d_in[7];

  const int Ntot = in_sizes[0] / 3;
  const int P3 = in_sizes[4];

  float* out_feat = (float*)d_out;
  float* out_imp = out_feat + (size_t)Ntot * FEAT;

  const size_t lds_bytes = (size_t)P3 * sizeof(float);
  const int block = 256;
  const int grid = 768;  // persistent blocks; 120KB LDS -> 2 blocks/WGP

  MultiLayerTetra_69664369541554_kernel<<<grid, block, lds_bytes, stream>>>(
      xyz, cell_id, old_cell_id, point_index, point_xyz, field, ili, thresh,
      out_feat, out_imp, Ntot, P3);
}